// ResidualBlock_22995254903322
// MI455X (gfx1250) — compile-verified
//
#include <hip/hip_runtime.h>

// ---------------------------------------------------------------------------
// ChebNet residual block on MI455X (gfx1250, wave32).
//  - SpMM: 1 edge per wave, contiguous (B*Cin) node rows, f32 L2 atomics,
//    fully unrolled scatter (R templated).
//  - Chebyshev recursion folded into scatter (init -x_{k-2}, scatter 2*w).
//  - Dense K-stacked einsum as one GEMM via V_WMMA_F32_16X16X4_F32 (full f32),
//    templated on (Cin, Cout); W tile staged in LDS (conflict-free ds_load),
//    A streamed as float2 global loads with compile-time strides.
//  - BatchNorm via LDS ds_add_f32 accumulators + global atomic flush.
// Workspace requirement: ~564 MB.
// ---------------------------------------------------------------------------

namespace {

constexpr int   V_NODES = 100000;
constexpr int   N_EDGES = 1600000;
constexpr int   BATCH   = 4;
constexpr int   M_ROWS  = V_NODES * BATCH;   // 400000 (v,b) rows
constexpr float EPS_BN  = 1e-5f;

typedef float v2f __attribute__((ext_vector_type(2)));
typedef float v8f __attribute__((ext_vector_type(8)));

// ---------------- elementwise helpers ----------------

__global__ __launch_bounds__(256) void fill_kernel(float* __restrict__ p, float v, size_t n) {
  size_t i = (size_t)blockIdx.x * blockDim.x + threadIdx.x;
  if (i < n) p[i] = v;
}

// dst = alpha * src   (alpha=1: copy, alpha=-1: negated init for cheb recursion)
__global__ __launch_bounds__(256) void scale_copy(float* __restrict__ dst,
                                                  const float* __restrict__ src,
                                                  float alpha, size_t n) {
  size_t i = (size_t)blockIdx.x * blockDim.x + threadIdx.x;
  if (i < n) dst[i] = alpha * src[i];
}

// ---------------- BatchNorm ----------------

// Stats over input layout (B, C, V): one (channel, chunk) per block, coalesced.
__global__ __launch_bounds__(256) void bn_stats_bcv(const float* __restrict__ X,
                                                    float* __restrict__ stats,
                                                    int V, int C, int chunks) {
  int c     = blockIdx.x / chunks;
  int chunk = blockIdx.x % chunks;
  float s = 0.f, ss = 0.f;
  for (int b = 0; b < BATCH; ++b) {
    const float* p = X + ((size_t)b * C + c) * (size_t)V;
    for (int v = chunk * blockDim.x + threadIdx.x; v < V; v += chunks * blockDim.x) {
      float x = p[v];
      s += x; ss += x * x;
    }
  }
  __shared__ float sh[512];
  sh[threadIdx.x] = s; sh[256 + threadIdx.x] = ss;
  __syncthreads();
  for (int o = blockDim.x >> 1; o > 0; o >>= 1) {
    if ((int)threadIdx.x < o) {
      sh[threadIdx.x]       += sh[threadIdx.x + o];
      sh[256 + threadIdx.x] += sh[256 + threadIdx.x + o];
    }
    __syncthreads();
  }
  if (threadIdx.x == 0) {
    atomicAdd(&stats[c],     sh[0]);
    atomicAdd(&stats[C + c], sh[256]);
  }
}

// Stats over node-major (M, C) layout with optional fused ReLU.
// C is 32/64 -> consecutive lanes hit distinct LDS channels (no ds_add conflicts).
__global__ __launch_bounds__(256) void bn_stats_mc(const float* __restrict__ X,
                                                   float* __restrict__ stats,
                                                   size_t n, int cmask, int C, int doRelu) {
  __shared__ float ls[128];
  for (int t = threadIdx.x; t < 2 * C; t += blockDim.x) ls[t] = 0.f;
  __syncthreads();
  size_t stride = (size_t)gridDim.x * blockDim.x;
  for (size_t i = (size_t)blockIdx.x * blockDim.x + threadIdx.x; i < n; i += stride) {
    float v = X[i];
    if (doRelu) v = fmaxf(v, 0.f);
    int c = (int)(i & (size_t)cmask);
    atomicAdd(&ls[c], v);
    atomicAdd(&ls[C + c], v * v);
  }
  __syncthreads();
  for (int t = threadIdx.x; t < 2 * C; t += blockDim.x) atomicAdd(&stats[t], ls[t]);
}

// scale = g * rsqrt(var + eps); shift = b - mean * scale
__global__ void bn_finalize(const float* __restrict__ stats,
                            const float* __restrict__ g, const float* __restrict__ b,
                            float* __restrict__ scsh, float invN, int C) {
  int c = threadIdx.x;
  if (c < C) {
    float mean = stats[c] * invN;
    float var  = stats[C + c] * invN - mean * mean;
    float sc   = g[c] * rsqrtf(var + EPS_BN);
    scsh[c]     = sc;
    scsh[C + c] = b[c] - mean * sc;
  }
}

// Apply BN to (B,C,V) input and transpose into node-major (M, C).
__global__ __launch_bounds__(256) void bn_apply_tr(const float* __restrict__ X,
                                                   const float* __restrict__ scsh,
                                                   float* __restrict__ Yt,
                                                   int V, int C, int lc) {
  size_t i = (size_t)blockIdx.x * blockDim.x + threadIdx.x;
  size_t total = (size_t)M_ROWS << lc;
  if (i >= total) return;
  int    c = (int)(i & (size_t)(C - 1));
  size_t m = i >> lc;
  int    b = (int)(m & 3);
  size_t v = m >> 2;
  float  x = X[((size_t)b * C + c) * (size_t)V + v];
  Yt[i] = scsh[c] * x + scsh[C + c];
}

// y = scale * relu(x) + shift  on node-major (M, C)
__global__ __launch_bounds__(256) void relu_bn_apply(float* __restrict__ dst,
                                                     const float* __restrict__ src,
                                                     const float* __restrict__ scsh,
                                                     int cmask, int C, size_t n) {
  size_t i = (size_t)blockIdx.x * blockDim.x + threadIdx.x;
  if (i >= n) return;
  int c = (int)(i & (size_t)cmask);
  dst[i] = scsh[c] * fmaxf(src[i], 0.f) + scsh[C + c];
}

// ---------------- SpMM scatter (1 edge per wave32) ----------------
// Y[row, :] += alpha * vals[e] * X[col, :], row length R = B*Cin (128 or 256).
template <int R>
__global__ __launch_bounds__(256) void spmm_scatter(const int* __restrict__ rows,
                                                    const int* __restrict__ cols,
                                                    const float* __restrict__ vals,
                                                    float alpha,
                                                    const float* __restrict__ X,
                                                    float* __restrict__ Y,
                                                    int E) {
  int wpb = blockDim.x >> 5;                       // waves per block (wave32)
  int e   = blockIdx.x * wpb + (threadIdx.x >> 5); // uniform per wave
  if (e >= E) return;
  int lane = threadIdx.x & 31;
  if (lane == 0 && e + 4096 < E) {                 // global_prefetch_b8 of edge stream
    __builtin_prefetch(cols + e + 4096, 0, 1);
  }
  int   r = rows[e];
  int   c = cols[e];
  float w = alpha * vals[e];
  const float* xs = X + (size_t)c * R + lane;
  float*       yd = Y + (size_t)r * R + lane;
#pragma unroll
  for (int i = 0; i < R / 32; ++i) {
    atomicAdd(yd + 32 * i, w * xs[32 * i]);        // global_atomic_add_f32 -> L2
  }
}

// ---------------- WMMA GEMM ----------------
// out(M,COUT) = sum_k Xk(M,CIN) @ W(k*CIN.., COUT) + bias
// Xc packed as (4, M, CIN); Wk is (4*CIN, COUT) row-major (= reference (K,Cin,Cout)).
// Block = 8 waves sharing one 16-wide output-channel tile (W tile staged in LDS);
// each wave owns one 16x16 tile, K-loop in steps of 4 via V_WMMA_F32_16X16X4_F32.
template <int CIN, int COUT, int LC>
__global__ __launch_bounds__(256) void cheb_gemm_wmma(const float* __restrict__ Xc,
                                                      const float* __restrict__ Wk,
                                                      const float* __restrict__ bias,
                                                      float* __restrict__ Yt) {
  constexpr int    KDIM    = 4 * CIN;
  constexpr int    TILES_N = COUT / 16;
  constexpr size_t KSTRIDE = (size_t)M_ROWS * CIN;   // cheb-order stride in Xc

  __shared__ float wtile[KDIM * 16];                 // 8 KB (CIN=32) / 16 KB (CIN=64)

  const int tn = blockIdx.x % TILES_N;               // output-channel tile (block-wide)
  const int gm = blockIdx.x / TILES_N;               // group of 8 row tiles
  const int o0 = tn * 16;

  // Cooperative, coalesced load of the W tile: rows 0..KDIM-1, cols o0..o0+15.
  for (int t = threadIdx.x; t < KDIM * 16; t += 256) {
    int krow = t >> 4, oc = t & 15;
    wtile[t] = Wk[(size_t)krow * COUT + (o0 + oc)];
  }
  __syncthreads();

  const int lane = threadIdx.x & 31;
  const int wv   = threadIdx.x >> 5;                 // wave in block: 0..7
  const int m0   = (gm * 8 + wv) * 16;               // this wave's row tile
  const int half = lane >> 4;                        // K sub-block select (ISA 16x4 A layout)
  const int l15  = lane & 15;
  const int boff = 32 * half + l15;                  // LDS B offset base (floats)

  v8f acc = {0.f, 0.f, 0.f, 0.f, 0.f, 0.f, 0.f, 0.f};
  const float* arow = Xc + (size_t)(m0 + l15) * CIN; // A row for this lane (M = lane&15)

#pragma unroll 8
  for (int kk = 0; kk < KDIM; kk += 4) {
    // kc = kk + 2*half never crosses a CIN boundary (kk%CIN <= CIN-4), so the
    // cheb order depends only on kk -> compile-time constant under unrolling.
    int ko = kk >> LC;
    int c  = (kk & (CIN - 1)) + 2 * half;
    v2f a  = *(const v2f*)(arow + (size_t)ko * KSTRIDE + c);  // global_load_b64
    v2f b;                                                    // 2x ds_load_b32, imm offsets
    b.x = wtile[kk * 16 + boff];
    b.y = wtile[kk * 16 + boff + 16];
    acc = __builtin_amdgcn_wmma_f32_16x16x4_f32(
        /*neg_a=*/false, a, /*neg_b=*/false, b,
        /*c_mod=*/(short)0, acc, /*reuse_a=*/false, /*reuse_b=*/false);
  }

  // D layout: VGPR r -> M=m0+r (lanes 0-15) / M=m0+8+r (lanes 16-31), N=o0+(lane&15)
  float bo = bias[o0 + l15];
  const int mD = m0 + 8 * half;
#pragma unroll
  for (int r = 0; r < 8; ++r) {
    Yt[(size_t)(mD + r) * COUT + (o0 + l15)] = acc[r] + bo;
  }
}

// ---------------- final residual epilogue: out(B,32,V) = relu(Ot + xn) transposed ----
__global__ __launch_bounds__(256) void epilogue_kernel(const float* __restrict__ Ot,
                                                       const float* __restrict__ Xn,
                                                       float* __restrict__ out, int V) {
  size_t i = (size_t)blockIdx.x * blockDim.x + threadIdx.x;
  size_t total = (size_t)BATCH * 32 * (size_t)V;
  if (i >= total) return;
  size_t v = i % (size_t)V;
  int    o = (int)((i / (size_t)V) & 31);
  int    b = (int)(i / ((size_t)32 * (size_t)V));
  size_t m = v * 4 + (size_t)b;
  float  val = Ot[m * 32 + o] + Xn[m * 32 + o];
  out[i] = fmaxf(val, 0.f);
}

}  // namespace

// ---------------------------------------------------------------------------

extern "C" void kernel_launch(void* const* d_in, const int* in_sizes, int n_in,
                              void* d_out, int out_size, void* d_ws, size_t ws_size,
                              hipStream_t stream) {
  (void)in_sizes; (void)n_in; (void)out_size; (void)ws_size;

  const float* x       = (const float*)d_in[0];
  const int*   ei      = (const int*)d_in[1];
  const float* vals    = (const float*)d_in[2];
  const float* in_bn_g = (const float*)d_in[3];
  const float* in_bn_b = (const float*)d_in[4];
  const float* in_w    = (const float*)d_in[5];
  const float* in_b    = (const float*)d_in[6];
  const float* h0_bn_g = (const float*)d_in[7];
  const float* h0_bn_b = (const float*)d_in[8];
  const float* h0_w    = (const float*)d_in[9];
  const float* h0_b    = (const float*)d_in[10];
  const float* h1_bn_g = (const float*)d_in[11];
  const float* h1_bn_b = (const float*)d_in[12];
  const float* h1_w    = (const float*)d_in[13];
  const float* h1_b    = (const float*)d_in[14];
  float* out = (float*)d_out;

  const int* rows = ei;
  const int* cols = ei + N_EDGES;

  // --- workspace carve (256B aligned): stats | scsh | xn | XC | OT ---
  char*  w     = (char*)d_ws;
  float* stats = (float*)w;                                    // 128 f
  float* scsh  = (float*)(w + 512);                            // 128 f
  float* xn    = (float*)(w + 1024);                           // M*32 f  (51.2 MB)
  float* XC    = xn + (size_t)M_ROWS * 32;                     // 4*M*64 f (409.6 MB)
  float* OT    = XC + (size_t)M_ROWS * 64 * 4;                 // M*64 f  (102.4 MB)

  auto g1 = [](size_t n) { return (unsigned)((n + 255) / 256); };
  const float invN = 1.f / (float)M_ROWS;
  const int spmm_blocks = N_EDGES / 8;                         // 8 waves/block, exact
  const int mgroups = (M_ROWS / 16) / 8;                       // 3125, exact

  // ---- input BN + transpose to node-major, keep xn for residual ----
  fill_kernel<<<1, 128, 0, stream>>>(stats, 0.f, 128);
  bn_stats_bcv<<<32 * 40, 256, 0, stream>>>(x, stats, V_NODES, 32, 40);
  bn_finalize<<<1, 64, 0, stream>>>(stats, in_bn_g, in_bn_b, scsh, invN, 32);
  bn_apply_tr<<<g1((size_t)M_ROWS * 32), 256, 0, stream>>>(x, scsh, xn, V_NODES, 32, 5);
  scale_copy<<<g1((size_t)M_ROWS * 32), 256, 0, stream>>>(XC, xn, 1.f, (size_t)M_ROWS * 32);

  // ---- chebyshev feature builder: x1=L x0 ; x2=2L x1-x0 ; x3=2L x2-x1 ----
  auto cheb = [&](auto cin_tag) {
    constexpr int CIN = decltype(cin_tag)::value;
    constexpr int R   = BATCH * CIN;           // node-row length
    size_t S = (size_t)M_ROWS * CIN;           // per-order stride
    float *x0 = XC, *x1 = XC + S, *x2 = XC + 2 * S, *x3 = XC + 3 * S;
    fill_kernel<<<g1(S), 256, 0, stream>>>(x1, 0.f, S);
    spmm_scatter<R><<<spmm_blocks, 256, 0, stream>>>(rows, cols, vals, 1.f, x0, x1, N_EDGES);
    scale_copy<<<g1(S), 256, 0, stream>>>(x2, x0, -1.f, S);
    spmm_scatter<R><<<spmm_blocks, 256, 0, stream>>>(rows, cols, vals, 2.f, x1, x2, N_EDGES);
    scale_copy<<<g1(S), 256, 0, stream>>>(x3, x1, -1.f, S);
    spmm_scatter<R><<<spmm_blocks, 256, 0, stream>>>(rows, cols, vals, 2.f, x2, x3, N_EDGES);
  };

  // ---- layer 0: cheb_conv 32 -> 64 ----
  cheb(std::integral_constant<int, 32>{});
  cheb_gemm_wmma<32, 64, 5><<<mgroups * 4, 256, 0, stream>>>(XC, in_w, in_b, OT);

  // ---- hidden 0: relu + BN + cheb_conv 64 -> 64 ----
  fill_kernel<<<1, 128, 0, stream>>>(stats, 0.f, 128);
  bn_stats_mc<<<2048, 256, 0, stream>>>(OT, stats, (size_t)M_ROWS * 64, 63, 64, 1);
  bn_finalize<<<1, 64, 0, stream>>>(stats, h0_bn_g, h0_bn_b, scsh, invN, 64);
  relu_bn_apply<<<g1((size_t)M_ROWS * 64), 256, 0, stream>>>(XC, OT, scsh, 63, 64,
                                                             (size_t)M_ROWS * 64);
  cheb(std::integral_constant<int, 64>{});
  cheb_gemm_wmma<64, 64, 6><<<mgroups * 4, 256, 0, stream>>>(XC, h0_w, h0_b, OT);

  // ---- hidden 1: relu + BN + cheb_conv 64 -> 32 ----
  fill_kernel<<<1, 128, 0, stream>>>(stats, 0.f, 128);
  bn_stats_mc<<<2048, 256, 0, stream>>>(OT, stats, (size_t)M_ROWS * 64, 63, 64, 1);
  bn_finalize<<<1, 64, 0, stream>>>(stats, h1_bn_g, h1_bn_b, scsh, invN, 64);
  relu_bn_apply<<<g1((size_t)M_ROWS * 64), 256, 0, stream>>>(XC, OT, scsh, 63, 64,
                                                             (size_t)M_ROWS * 64);
  cheb(std::integral_constant<int, 64>{});
  cheb_gemm_wmma<64, 32, 6><<<mgroups * 2, 256, 0, stream>>>(XC, h1_w, h1_b, OT);

  // ---- residual + relu + transpose back to (B, 32, V) ----
  epilogue_kernel<<<g1((size_t)M_ROWS * 32), 256, 0, stream>>>(OT, xn, out, V_NODES);
}